// StrainEnergyPotential_41403484733959
// MI455X (gfx1250) — compile-verified
//
#include <hip/hip_runtime.h>
#include <stdint.h>

// ---------------------------------------------------------------------------
// StrainEnergyPotential gradient: out = dPsi/dE(E) - dPsi/dE(0), elementwise
// over 16.8M 3-vectors. Memory-bound (402 MB @ 23.3 TB/s ~= 17 us floor).
// CDNA5 path: double-buffered global_load_async_to_lds_b128 staging
// (ASYNCcnt split-wait pipelining) + ds_load_b128 + nontemporal b128 stores.
// wave32 tiles of 128 points (1536 B).
// ---------------------------------------------------------------------------

#define THREADS 256
#define WAVES_PER_BLOCK (THREADS / 32)
#define PTS_PER_LANE 4
#define PTS_PER_WAVE (32 * PTS_PER_LANE)       // 128 points per wave tile
#define FLOATS_PER_WAVE (PTS_PER_WAVE * 3)     // 384 floats
#define BYTES_PER_WAVE (FLOATS_PER_WAVE * 4)   // 1536 bytes

typedef __attribute__((ext_vector_type(4))) float v4f;

struct PsiParams {
    float w00, w01, w10, w11;
    float a0, a1;
    float gz;  // 2*(dPsi/dI1|0 + dPsi/dI2|0), the constant baseline gradient
};

__device__ __forceinline__ void psi_grad_point(const PsiParams& P,
                                               float e0, float e1, float e2,
                                               float& o0, float& o1, float& o2) {
    // C = 2E + diag-shift (Voigt)
    float c11 = __builtin_fmaf(2.0f, e0, 1.0f);
    float c22 = __builtin_fmaf(2.0f, e1, 1.0f);
    float c12 = 2.0f * e2;
    float I1 = c11 + c22;
    float I2 = __builtin_fmaf(-0.25f * c12, c12, c11 * c22);
    // P_i = exp(W[i,0]*ln I1 + W[i,1]*ln I2)  (hardware trans ops)
    float l1 = __logf(I1);
    float l2 = __logf(I2);
    float p0 = __expf(__builtin_fmaf(P.w00, l1, P.w01 * l2));
    float p1 = __expf(__builtin_fmaf(P.w10, l1, P.w11 * l2));
    float dI1 = 0.5f * (P.a0 * P.w00 * p0 + P.a1 * P.w10 * p1) * __builtin_amdgcn_rcpf(I1);
    float dI2 = 0.5f * (P.a0 * P.w01 * p0 + P.a1 * P.w11 * p1) * __builtin_amdgcn_rcpf(I2);
    o0 = 2.0f * __builtin_fmaf(dI2, c22, dI1) - P.gz;
    o1 = 2.0f * __builtin_fmaf(dI2, c11, dI1) - P.gz;
    o2 = -c12 * dI2;
}

// Issue one 1536B wave-tile (3 x 512B) of async global->LDS b128 copies.
// INST_OFFSET applies to both the LDS and the global address (GVS mode).
__device__ __forceinline__ void async_load_tile(unsigned ldsA, unsigned gOff,
                                                const float* base) {
    asm volatile(
        "global_load_async_to_lds_b128 %0, %1, %2\n\t"
        "global_load_async_to_lds_b128 %0, %1, %2 offset:512\n\t"
        "global_load_async_to_lds_b128 %0, %1, %2 offset:1024"
        :
        : "v"(ldsA), "v"(gOff), "s"(base)
        : "memory");
}

__device__ __forceinline__ void compute_store_tile(const PsiParams& P,
                                                   const float* ldsTile, int lane,
                                                   float* __restrict__ out, int tile) {
    // Each lane reads back its own 48B (4 AoS points) -> 3 x ds_load_b128.
    const v4f* s4 = (const v4f*)(ldsTile + lane * 12);
    v4f A = s4[0], Bv = s4[1], Cv = s4[2];
    float in[12] = {A.x,  A.y,  A.z,  A.w,
                    Bv.x, Bv.y, Bv.z, Bv.w,
                    Cv.x, Cv.y, Cv.z, Cv.w};
    float o[12];
#pragma unroll
    for (int p = 0; p < 4; ++p) {
        psi_grad_point(P, in[3 * p], in[3 * p + 1], in[3 * p + 2],
                       o[3 * p], o[3 * p + 1], o[3 * p + 2]);
    }
    // Output is write-once: non-temporal b128 stores, bypass RT caching.
    float* dst = out + (size_t)tile * FLOATS_PER_WAVE + lane * 12;
    v4f O0 = {o[0], o[1], o[2],  o[3]};
    v4f O1 = {o[4], o[5], o[6],  o[7]};
    v4f O2 = {o[8], o[9], o[10], o[11]};
    __builtin_nontemporal_store(O0, (v4f*)(dst + 0));
    __builtin_nontemporal_store(O1, (v4f*)(dst + 4));
    __builtin_nontemporal_store(O2, (v4f*)(dst + 8));
}

__global__ __launch_bounds__(THREADS)
void StrainEnergyGrad_kernel(const float* __restrict__ E,
                             const float* __restrict__ W,
                             const float* __restrict__ ab,
                             float* __restrict__ out,
                             int n_points) {
    // Two staging buffers per wave: software pipeline over ASYNCcnt.
    __shared__ float buf[WAVES_PER_BLOCK][2][FLOATS_PER_WAVE];

    const int lane      = threadIdx.x & 31;
    const int waveInBlk = threadIdx.x >> 5;
    const int waveGlob  = blockIdx.x * WAVES_PER_BLOCK + waveInBlk;
    const int waveTotal = gridDim.x * WAVES_PER_BLOCK;

    // Tiny uniform parameters -> scalar float ALU; baseline gradient at E=0
    // (I1=2, I2=1 => P_i|0 = 2^{W[i,0]}).  d_e never affects the gradient.
    PsiParams P;
    P.w00 = W[0]; P.w01 = W[1]; P.w10 = W[2]; P.w11 = W[3];
    P.a0  = ab[0]; P.a1 = ab[1];
    {
        float q0 = exp2f(P.w00), q1 = exp2f(P.w10);
        float dI1_0 = 0.25f * (P.a0 * P.w00 * q0 + P.a1 * P.w10 * q1);
        float dI2_0 = 0.50f * (P.a0 * P.w01 * q0 + P.a1 * P.w11 * q1);
        P.gz = 2.0f * (dI1_0 + dI2_0);
    }

    const int numTiles = n_points / PTS_PER_WAVE;

    // 32-bit LDS byte addresses of this wave's two staging buffers (low 32
    // bits of the flat LDS address == DS offset on gfx1250).
    unsigned ldsA0 = (unsigned)(uintptr_t)(&buf[waveInBlk][0][0]) + (unsigned)lane * 16u;
    unsigned ldsA1 = (unsigned)(uintptr_t)(&buf[waveInBlk][1][0]) + (unsigned)lane * 16u;

    int tile = waveGlob;
    int cur = 0;
    if (tile < numTiles) {
        async_load_tile(ldsA0,
                        (unsigned)tile * (unsigned)BYTES_PER_WAVE + (unsigned)lane * 16u, E);
    }
    for (; tile < numTiles; tile += waveTotal, cur ^= 1) {
        const int next = tile + waveTotal;
        if (next < numTiles) {
            // Prefetch the next tile into the alternate buffer, then wait for
            // the OLDER group of 3 only (async loads complete in order):
            // ASYNCcnt <= 3 keeps the new 3 in flight while we compute.
            async_load_tile(cur ? ldsA0 : ldsA1,
                            (unsigned)next * (unsigned)BYTES_PER_WAVE + (unsigned)lane * 16u, E);
            asm volatile("s_wait_asynccnt 0x3" ::: "memory");
        } else {
            asm volatile("s_wait_asynccnt 0x0" ::: "memory");
        }
        compute_store_tile(P, &buf[waveInBlk][cur][0], lane, out, tile);
    }

    // Scalar tail for n_points not a multiple of 128 (unused for 4096x4096).
    const int done = numTiles * PTS_PER_WAVE;
    const int gtid = blockIdx.x * THREADS + threadIdx.x;
    const int nthr = gridDim.x * THREADS;
    for (int p = done + gtid; p < n_points; p += nthr) {
        float o0, o1, o2;
        psi_grad_point(P, E[3 * p], E[3 * p + 1], E[3 * p + 2], o0, o1, o2);
        out[3 * p]     = o0;
        out[3 * p + 1] = o1;
        out[3 * p + 2] = o2;
    }
}

extern "C" void kernel_launch(void* const* d_in, const int* in_sizes, int n_in,
                              void* d_out, int out_size, void* d_ws, size_t ws_size,
                              hipStream_t stream) {
    (void)n_in; (void)out_size; (void)d_ws; (void)ws_size;
    const float* E  = (const float*)d_in[0];   // (B, S, 3) f32
    const float* W  = (const float*)d_in[1];   // (2, 2)    f32
    const float* ab = (const float*)d_in[2];   // (2,)      f32
    // d_in[3] = d_e: additive constant in psi, gradient-independent.
    float* out = (float*)d_out;

    const int n_total  = in_sizes[0];
    const int n_points = n_total / 3;
    const int numTiles = n_points / PTS_PER_WAVE;

    int blocks = numTiles / WAVES_PER_BLOCK;   // grid-stride: ~8 tiles/wave
    if (blocks > 2048) blocks = 2048;
    if (blocks < 1) blocks = 1;

    StrainEnergyGrad_kernel<<<blocks, THREADS, 0, stream>>>(E, W, ab, out, n_points);
}